// PowerIterationConv_45595372814669
// MI455X (gfx1250) — compile-verified
//
#include <hip/hip_runtime.h>

// ---------------------------------------------------------------------------
// Gram-iteration spectral norm (Delattre 2024) for a 192x192x3x3 conv kernel.
// All heavy math runs through V_WMMA_F32_16X16X4_F32 (full fp32 precision).
// ---------------------------------------------------------------------------

typedef float v2f __attribute__((ext_vector_type(2)));
typedef float v8f __attribute__((ext_vector_type(8)));

#define C_CH 192
#define N_IN (192 * 192 * 9)

__device__ __forceinline__ v8f wmma_f32_16x16x4(v2f a, v2f b, v8f c) {
  // 8 args: (neg_a, A, neg_b, B, c_mod, C, reuse_a, reuse_b)
  return __builtin_amdgcn_wmma_f32_16x16x4_f32(false, a, false, b, (short)0, c,
                                               false, false);
}

// [o][a][p] (input layout) -> [o][p][a] (GEMM-friendly: channel fastest)
__global__ void transpose_in_k(const float* __restrict__ in,
                               float* __restrict__ out, int n) {
  int i = blockIdx.x * blockDim.x + threadIdx.x;
  if (i < n) {
    int a = i % C_CH;
    int t = i / C_CH;  // t = o*9 + p
    int p = t % 9;
    int o = t / 9;
    out[i] = in[(o * C_CH + a) * 9 + p];
  }
}

__global__ void ssq_partial_k(const float* __restrict__ x, int n,
                              float* __restrict__ part) {
  __shared__ float sdata[256];
  float s = 0.f;
  for (int i = blockIdx.x * blockDim.x + threadIdx.x; i < n;
       i += gridDim.x * blockDim.x) {
    float v = x[i];
    s += v * v;
  }
  sdata[threadIdx.x] = s;
  __syncthreads();
  for (int off = 128; off > 0; off >>= 1) {
    if ((int)threadIdx.x < off) sdata[threadIdx.x] += sdata[threadIdx.x + off];
    __syncthreads();
  }
  if (threadIdx.x == 0) part[blockIdx.x] = sdata[0];
}

__global__ void final_reduce_k(const float* __restrict__ part, int n,
                               float* __restrict__ out) {
  __shared__ float sdata[256];
  float s = 0.f;
  for (int i = threadIdx.x; i < n; i += 256) s += part[i];
  sdata[threadIdx.x] = s;
  __syncthreads();
  for (int off = 128; off > 0; off >>= 1) {
    if ((int)threadIdx.x < off) sdata[threadIdx.x] += sdata[threadIdx.x + off];
    __syncthreads();
  }
  if (threadIdx.x == 0) out[0] = sdata[0];
}

// Gram conv via per-shift GEMM: one wave computes a 32x64 tile of
//   G_s[a,b] = (1/f^2) * sum_{o,p} src[o, p + s, a] * src[o, p, b]
// src layout: [o][p][c] with p < k*k (row-major k x k), c = 192.
// dst layout: [a][s][b] (i.e. already transposed for the next iteration).
// If dst == nullptr: instead accumulate sum of squares of the (scaled) tile
// into part[block] (fused final Frobenius norm).
__global__ __launch_bounds__(32) void gram_wmma_k(
    const float* __restrict__ src, float* __restrict__ dst,
    const float* __restrict__ f2, float* __restrict__ part, int k) {
  const int P = k * k;
  const int w = 2 * k - 1;
  const int S = w * w;
  const int sidx = blockIdx.y;
  const int sy = sidx / w - (k - 1);
  const int sx = sidx % w - (k - 1);
  const int ny = k - (sy < 0 ? -sy : sy);
  const int nx = k - (sx < 0 ? -sx : sx);
  const int validp = ny * nx;  // overlap positions (>= 1)
  const int by0 = sy < 0 ? -sy : 0;
  const int bx0 = sx < 0 ? -sx : 0;
  const int dpa = sy * k + sx;  // pa - pb is shift-constant
  const unsigned Mdiv = (65536u + (unsigned)nx - 1u) / (unsigned)nx;  // j/nx

  const int lane = threadIdx.x;
  const int l15 = lane & 15;
  const int h = lane >> 4;               // selects kk = {2h, 2h+1}
  const int m0 = (blockIdx.x / 3) * 32;  // a-tile base (6 tiles of 32)
  const int n0 = (blockIdx.x % 3) * 64;  // b-tile base (3 tiles of 64)

  const v8f z = {0.f, 0.f, 0.f, 0.f, 0.f, 0.f, 0.f, 0.f};
  v8f acc[2][4] = {{z, z, z, z}, {z, z, z, z}};

  for (int o = 0; o < C_CH; ++o) {
    const float* baseo = src + (size_t)o * P * C_CH;
    if (o + 1 < C_CH) __builtin_prefetch(baseo + P * C_CH, 0, 3);

    int j0 = 0;
    // ---- main body: no masking, no clamping (j0+3 < validp) ----
    for (; j0 + 4 <= validp; j0 += 4) {
      int jb = j0 + 2 * h;
      int jd0 = (int)(((unsigned)jb * Mdiv) >> 16);  // jb / nx (exact)
      int jm0 = jb - jd0 * nx;
      int jm1 = jm0 + 1;
      bool wrap = (jm1 == nx);
      int jd1 = jd0 + (wrap ? 1 : 0);
      jm1 = wrap ? 0 : jm1;
      int pb0 = (by0 + jd0) * k + (bx0 + jm0);
      int pb1 = (by0 + jd1) * k + (bx0 + jm1);
      const float* rb0 = baseo + pb0 * C_CH;
      const float* rb1 = baseo + pb1 * C_CH;
      const float* ra0 = rb0 + dpa * C_CH;
      const float* ra1 = rb1 + dpa * C_CH;

      v2f av[2], bv[4];
      av[0][0] = ra0[m0 + l15];
      av[1][0] = ra0[m0 + 16 + l15];
      av[0][1] = ra1[m0 + l15];
      av[1][1] = ra1[m0 + 16 + l15];
#pragma unroll
      for (int tj = 0; tj < 4; ++tj) {
        bv[tj][0] = rb0[n0 + 16 * tj + l15];
        bv[tj][1] = rb1[n0 + 16 * tj + l15];
      }
#pragma unroll
      for (int ti = 0; ti < 2; ++ti)
#pragma unroll
        for (int tj = 0; tj < 4; ++tj)
          acc[ti][tj] = wmma_f32_16x16x4(av[ti], bv[tj], acc[ti][tj]);
    }
    // ---- masked tail (uniform branch; at most one step) ----
    if (j0 < validp) {
      v2f av[2], bv[4];
#pragma unroll
      for (int e = 0; e < 2; ++e) {
        int j = j0 + 2 * h + e;
        int jj = j < validp ? j : validp - 1;
        int jd = (int)(((unsigned)jj * Mdiv) >> 16);
        int jm = jj - jd * nx;
        int pb = (by0 + jd) * k + (bx0 + jm);
        const float* rb = baseo + pb * C_CH;
        const float* ra = rb + dpa * C_CH;
        bool ok = j < validp;
        av[0][e] = ok ? ra[m0 + l15] : 0.f;
        av[1][e] = ok ? ra[m0 + 16 + l15] : 0.f;
#pragma unroll
        for (int tj = 0; tj < 4; ++tj)
          bv[tj][e] = ok ? rb[n0 + 16 * tj + l15] : 0.f;
      }
#pragma unroll
      for (int ti = 0; ti < 2; ++ti)
#pragma unroll
        for (int tj = 0; tj < 4; ++tj)
          acc[ti][tj] = wmma_f32_16x16x4(av[ti], bv[tj], acc[ti][tj]);
    }
  }

  const float inv = 1.0f / f2[0];  // gram(K/f) = gram(K) / f^2

  if (dst) {
    // D layout: VGPR r -> M = r + 8*h, N = l15
#pragma unroll
    for (int ti = 0; ti < 2; ++ti)
#pragma unroll
      for (int r = 0; r < 8; ++r) {
        int am = m0 + 16 * ti + r + 8 * h;
        size_t row = ((size_t)am * S + sidx) * C_CH;
#pragma unroll
        for (int tj = 0; tj < 4; ++tj)
          dst[row + n0 + 16 * tj + l15] = acc[ti][tj][r] * inv;
      }
  } else {
    float ssq = 0.f;
#pragma unroll
    for (int ti = 0; ti < 2; ++ti)
#pragma unroll
      for (int tj = 0; tj < 4; ++tj)
#pragma unroll
        for (int r = 0; r < 8; ++r) {
          float g = acc[ti][tj][r] * inv;
          ssq += g * g;
        }
    for (int off = 16; off > 0; off >>= 1) ssq += __shfl_xor(ssq, off, 32);
    if (lane == 0) part[blockIdx.y * gridDim.x + blockIdx.x] = ssq;
  }
}

// sigma = exp( sum_t log(f_t)/2^t ), f_t = sqrt(S_t) -> 0.5*log(S_t)
__global__ void sigma_k(const float* __restrict__ s,
                        float* __restrict__ inv_sigma) {
  if (threadIdx.x == 0 && blockIdx.x == 0) {
    double ls = 0.5 * (log((double)s[0]) + 0.5 * log((double)s[1]) +
                       0.25 * log((double)s[2]) + 0.125 * log((double)s[3]));
    inv_sigma[0] = (float)exp(-ls);
  }
}

__global__ void scale_k(const float* __restrict__ in,
                        const float* __restrict__ inv_sigma,
                        float* __restrict__ out, int n) {
  int i = blockIdx.x * blockDim.x + threadIdx.x;
  if (i < n) out[i] = in[i] * inv_sigma[0];
}

static inline int imin(int a, int b) { return a < b ? a : b; }

extern "C" void kernel_launch(void* const* d_in, const int* in_sizes, int n_in,
                              void* d_out, int out_size, void* d_ws,
                              size_t ws_size, hipStream_t stream) {
  (void)in_sizes; (void)n_in; (void)out_size; (void)ws_size;
  const float* K = (const float*)d_in[0];
  float* out = (float*)d_out;
  char* ws = (char*)d_ws;

  float* scal = (float*)ws;               // [0..3]=S0..S3, [4]=1/sigma
  float* part = (float*)(ws + 1024);      // up to 16384 partials (64 KB)
  float* K1T = (float*)(ws + (1 << 17));  // 192*192*25 f32 = 3.69 MB
  float* K2T = (float*)(ws + (4 << 20));  // 192*192*81 f32 = 11.94 MB
  float* KT0 = K2T;                       // t0 transposed input (dead by t1)

  const int N0 = N_IN;            // 331776
  const int N1 = 192 * 192 * 25;  // 921600
  const int N2 = 192 * 192 * 81;  // 2985984

  // t=0 prep: transpose input, S0 = ||K||_F^2
  transpose_in_k<<<(N0 + 255) / 256, 256, 0, stream>>>(K, KT0, N0);
  int g0 = imin(1024, (N0 + 255) / 256);
  ssq_partial_k<<<g0, 256, 0, stream>>>(K, N0, part);
  final_reduce_k<<<1, 256, 0, stream>>>(part, g0, scal + 0);

  // t=0 gram: (3x3) -> K1 (5x5), scaled by 1/S0, stored transposed
  gram_wmma_k<<<dim3(18, 25), 32, 0, stream>>>(KT0, K1T, scal + 0, nullptr, 3);

  // S1 = ||K1||_F^2
  int g1 = imin(1024, (N1 + 255) / 256);
  ssq_partial_k<<<g1, 256, 0, stream>>>(K1T, N1, part);
  final_reduce_k<<<1, 256, 0, stream>>>(part, g1, scal + 1);

  // t=1 gram: (5x5) -> K2 (9x9)
  gram_wmma_k<<<dim3(18, 81), 32, 0, stream>>>(K1T, K2T, scal + 1, nullptr, 5);

  // S2 = ||K2||_F^2
  int g2 = imin(1024, (N2 + 255) / 256);
  ssq_partial_k<<<g2, 256, 0, stream>>>(K2T, N2, part);
  final_reduce_k<<<1, 256, 0, stream>>>(part, g2, scal + 2);

  // t=2 gram (9x9 -> 17x17): never materialized; fused ||K3||_F^2
  gram_wmma_k<<<dim3(18, 289), 32, 0, stream>>>(K2T, nullptr, scal + 2, part,
                                                9);
  final_reduce_k<<<1, 256, 0, stream>>>(part, 18 * 289, scal + 3);

  // sigma and final scaling of the ORIGINAL kernel
  sigma_k<<<1, 1, 0, stream>>>(scal, scal + 4);
  scale_k<<<(N0 + 255) / 256, 256, 0, stream>>>(K, scal + 4, out, N0);
}